// QuantizedModule_90589450207296
// MI455X (gfx1250) — compile-verified
//
#include <hip/hip_runtime.h>
#include <hip/hip_bf16.h>
#include <stdint.h>

// Blockwise int8 dequantization (bitsandbytes-style):
//   out[i] = code[weight[i]] * absmax[i / 4096]
//
// Memory-bound: 256 MiB codes in, 256 MiB floats out -> ~23 us floor at
// 23.3 TB/s. Strategy:
//   - 1 workgroup per 4096-element quant block; absmax is a scalar load.
//   - 256-entry codebook staged into LDS once per workgroup via the
//     CDNA5 async-to-LDS path (global_load_async_to_lds_b32 + s_wait_asynccnt).
//   - Streamed codes/output use non-temporal B128 loads/stores so the
//     512 MiB stream doesn't churn L2.

#define QBLOCK 4096

typedef int   v4i __attribute__((ext_vector_type(4)));
typedef float v4f __attribute__((ext_vector_type(4)));

__global__ __launch_bounds__(256) void dequant_blockwise_kernel(
    const int* __restrict__ codes,      // [O*I] int codes in [0,256)
    const float* __restrict__ absmax,   // [n_blocks]
    const float* __restrict__ codebook, // [256]
    float* __restrict__ out)            // [O*I]
{
    __shared__ float lut[256];

    const int tid = threadIdx.x;

    // ---- Stage the 256-entry codebook into LDS via CDNA5 async copy ----
    // One dword per thread (256 threads -> whole table in one shot).
    {
        const float* gsrc = codebook + tid;
        // Low 32 bits of the flat shared-pointer are the LDS byte offset.
        unsigned lds_byte = (unsigned)(uintptr_t)(&lut[tid]);
        asm volatile("global_load_async_to_lds_b32 %0, %1, off"
                     :
                     : "v"(lds_byte), "v"(gsrc)
                     : "memory");
        asm volatile("s_wait_asynccnt 0" ::: "memory");
    }
    __syncthreads();

    // ---- One quant block (4096 contiguous elements) per workgroup ----
    const size_t blk  = (size_t)blockIdx.x;
    const float scale = absmax[blk];          // uniform -> scalar load
    const size_t base = blk * (size_t)QBLOCK;

    // 4 passes; each pass: 256 threads x v4i = 1024 contiguous ints.
#pragma unroll
    for (int p = 0; p < 4; ++p) {
        const size_t idx = base + (size_t)p * 1024u + (size_t)tid * 4u;

        const v4i c = __builtin_nontemporal_load((const v4i*)(codes + idx));

        v4f r;
        r.x = lut[c.x & 255] * scale;
        r.y = lut[c.y & 255] * scale;
        r.z = lut[c.z & 255] * scale;
        r.w = lut[c.w & 255] * scale;

        __builtin_nontemporal_store(r, (v4f*)(out + idx));
    }
}

extern "C" void kernel_launch(void* const* d_in, const int* in_sizes, int n_in,
                              void* d_out, int out_size, void* d_ws, size_t ws_size,
                              hipStream_t stream) {
    const int*   codes    = (const int*)d_in[0];    // quant_weight [O,I] int32
    const float* absmax   = (const float*)d_in[1];  // [n_blocks]
    const float* codebook = (const float*)d_in[2];  // [256]
    float*       out      = (float*)d_out;          // [O,I] float32

    const int n_blocks = in_sizes[1];               // (O*I)/4096 = 16384

    dequant_blockwise_kernel<<<n_blocks, 256, 0, stream>>>(codes, absmax, codebook, out);
}